// AggregationLayer_20641612825103
// MI455X (gfx1250) — compile-verified
//
#include <hip/hip_runtime.h>
#include <math.h>

// Problem constants (from reference): B=64, L=2048, U=D=128, C=16
#define BB 64
#define LL 2048
#define UU 128
#define DD 128
#define CCLS 16

typedef float v2f __attribute__((ext_vector_type(2)));
typedef float v8f __attribute__((ext_vector_type(8)));
typedef unsigned int u32x4 __attribute__((ext_vector_type(4)));
typedef int i32x4 __attribute__((ext_vector_type(4)));
typedef int i32x8 __attribute__((ext_vector_type(8)));

// ---------------------------------------------------------------------------
// f32 WMMA: D(16x16) = A(16x4) * B(4x16) + C   (CDNA5 V_WMMA_F32_16X16X4_F32)
// ---------------------------------------------------------------------------
__device__ __forceinline__ v8f wmma4(v2f a, v2f b, v8f c) {
  return __builtin_amdgcn_wmma_f32_16x16x4_f32(false, a, false, b, (short)0, c,
                                               false, false);
}

// A fragment (16x4, f32) from an LDS tile with 132-float padded rows.
// Lane t holds M = t&15; element j holds K = kb*4 + (t>>4)*2 + j.
__device__ __forceinline__ v2f ldA_lds(const float* s, int lane, int kb) {
  int m  = lane & 15;
  int k0 = (kb << 2) + ((lane >> 4) << 1);
  v2f a;
  a.x = s[m * 132 + k0];
  a.y = s[m * 132 + k0 + 1];
  return a;
}

// B fragment (4x16, f32) from global row-major weights (K x N, ld = N).
__device__ __forceinline__ v2f ldB_g(const float* __restrict__ w, int ld,
                                     int col0, int lane, int kb) {
  int n  = col0 + (lane & 15);
  int k0 = (kb << 2) + ((lane >> 4) << 1);
  v2f b;
  b.x = w[k0 * ld + n];
  b.y = w[(k0 + 1) * ld + n];
  return b;
}

__device__ __forceinline__ float fsigmoid(float x) {
  return 1.0f / (1.0f + __expf(-x));
}
__device__ __forceinline__ float ftanh(float x) {
  float t = __expf(-2.0f * x);
  return (1.0f - t) / (1.0f + t);
}

// ---------------------------------------------------------------------------
// TDM: DMA a (rows x 128) f32 tile from global into LDS with a 4-DWORD pad
// every 128 DWORDs (=> 132-float padded rows, matching ldA_lds).
// D# packing per CDNA5 ISA 8.3/8.4 (group0: count/lds/global/type;
// group1: data_size=4B, pad_enable, pad_interval=6 (128 dw), pad_amount=3
// (4 dw), tensor/tile dims, dim0 stride).  2-D tile; groups 2/3 unused.
// This toolchain exposes the 6-arg builtin form (extra i32x8 group).
// ---------------------------------------------------------------------------
__device__ __forceinline__ void tdm_load_tile_f32(const float* gsrc,
                                                  unsigned lds_byte_addr,
                                                  unsigned rows) {
  unsigned long long ga = (unsigned long long)(size_t)gsrc;
  u32x4 g0;
  g0[0] = 1u;                                   // count=1, user descriptor
  g0[1] = lds_byte_addr;                        // lds_addr (bytes)
  g0[2] = (unsigned)(ga & 0xFFFFFFFFull);       // global_addr[31:0]
  g0[3] = (unsigned)((ga >> 32) & 0x1FFFFFFull) // global_addr[56:32]
          | (2u << 30);                         // type = 2 ("image")
  i32x8 g1;
  g1[0] = (int)((2u << 16)      // data_size = 4 bytes
                | (1u << 20)    // pad_enable
                | (6u << 22)    // pad_interval: 128 DWORDs
                | (3u << 25));  // pad_amount:   4 DWORDs
  g1[1] = (int)(128u << 16);                    // tensor_dim0 = 128 (lo16)
  g1[2] = (int)(rows << 16);                    // tensor_dim0 hi | tensor_dim1
  g1[3] = (int)(128u << 16);                    // tensor_dim1 hi | tile_dim0=128
  g1[4] = (int)rows;                            // tile_dim1 | tile_dim2=0
  g1[5] = 128;                                  // tensor_dim0_stride = 128
  g1[6] = 0;
  g1[7] = 0;
  i32x4 z4 = {0, 0, 0, 0};
  i32x8 z8 = {0, 0, 0, 0, 0, 0, 0, 0};
  __builtin_amdgcn_tensor_load_to_lds(g0, g1, z4, z4, z8, 0);
}

// ---------------------------------------------------------------------------
// Kernel 1: TreeLSTM gates + logits.  One wave32 per 16-row tile of (B*L).
// ---------------------------------------------------------------------------
__global__ __launch_bounds__(32) void tree_lstm_logits_kernel(
    const float* __restrict__ trip,    // (B, L, 128)
    const float* __restrict__ region,  // (B, L+1, 128)
    const float* __restrict__ W,       // (128, 512)
    const float* __restrict__ Uiou,    // (128, 384)
    const float* __restrict__ Uf,      // (128, 128)
    const float* __restrict__ cc,      // (2, 128)
    const float* __restrict__ Wcls,    // (128, 16)
    float* __restrict__ logits)        // (B, L, 16)
{
  __shared__ float sTrip[16 * 132];  // trip tile, padded rows
  __shared__ float sReg[17 * 132];   // region rows l0..l0+16 (hl + hr share)
  __shared__ float sHt[16 * 20];     // h-slice C->A transpose buffer

  const int lane = threadIdx.x;
  const long r0  = (long)blockIdx.x * 16;   // first global row (b*L + l)
  const int  b   = (int)(r0 >> 11);         // / L
  const int  l0  = (int)(r0 & (LL - 1));    // % L (tiles never cross batch)

  // ---- Stage tiles via Tensor Data Mover (async DMA -> padded LDS) ----
  const float* tripBase = trip + r0 * DD;
  const float* regBase  = region + ((long)b * (LL + 1) + l0) * UU;
  tdm_load_tile_f32(tripBase, (unsigned)(size_t)(void*)sTrip, 16u);
  tdm_load_tile_f32(regBase, (unsigned)(size_t)(void*)sReg, 17u);
  __builtin_amdgcn_s_wait_tensorcnt(0);

  const float* st  = sTrip;
  const float* slA = sReg;        // hl rows: l0 .. l0+15
  const float* srA = sReg + 132;  // hr rows: l0+1 .. l0+16

  const int nlane = lane & 15;
  const int mhalf = (lane >> 4) << 3;  // 0 or 8 (C-fragment row offset)

  v8f logacc = {};  // persistent logits accumulator (16x16)

  // ---- 8 N-blocks of 16 columns over U=128 ----
  for (int n = 0; n < 8; ++n) {
    const int n0 = n << 4;
    v8f ai = {}, af = {}, ao = {}, au = {};
    v8f ri = {}, ro = {}, ru = {};
    v8f afl = {}, afr = {};

#pragma unroll 2
    for (int kb = 0; kb < 32; ++kb) {
      v2f aT = ldA_lds(st, lane, kb);
      v2f aL = ldA_lds(slA, lane, kb);
      v2f aR = ldA_lds(srA, lane, kb);
      v2f aS;
      aS.x = aL.x + aR.x;
      aS.y = aL.y + aR.y;

      v2f bi = ldB_g(W, 4 * UU, 0 * UU + n0, lane, kb);
      v2f bf = ldB_g(W, 4 * UU, 1 * UU + n0, lane, kb);
      v2f bo = ldB_g(W, 4 * UU, 2 * UU + n0, lane, kb);
      v2f bu = ldB_g(W, 4 * UU, 3 * UU + n0, lane, kb);
      v2f ui = ldB_g(Uiou, 3 * UU, 0 * UU + n0, lane, kb);
      v2f uo = ldB_g(Uiou, 3 * UU, 1 * UU + n0, lane, kb);
      v2f uu = ldB_g(Uiou, 3 * UU, 2 * UU + n0, lane, kb);
      v2f uf = ldB_g(Uf, UU, n0, lane, kb);

      ai  = wmma4(aT, bi, ai);   // xi
      af  = wmma4(aT, bf, af);   // xf
      ao  = wmma4(aT, bo, ao);   // xo
      au  = wmma4(aT, bu, au);   // xu
      ri  = wmma4(aS, ui, ri);   // ri
      ro  = wmma4(aS, uo, ro);   // ro
      ru  = wmma4(aS, uu, ru);   // ru
      afl = wmma4(aL, uf, afl);  // hl @ U_f
      afr = wmma4(aR, uf, afr);  // hr @ U_f
    }

    // ---- Fused elementwise gates in C-fragment layout ----
    const float ccl = cc[n0 + nlane];        // cell_children[0]
    const float ccr = cc[UU + n0 + nlane];   // cell_children[1]
#pragma unroll
    for (int r = 0; r < 8; ++r) {
      float iv  = fsigmoid(ai[r] + ri[r]);
      float ov  = fsigmoid(ao[r] + ro[r]);
      float uv  = ftanh(au[r] + ru[r]);
      float flv = fsigmoid(af[r] + afl[r]);
      float frv = fsigmoid(af[r] + afr[r]);
      float cv  = iv * uv + flv * ccl + frv * ccr;
      float hv  = ov * ftanh(cv);
      sHt[(r + mhalf) * 20 + nlane] = hv;  // C-layout -> LDS (padded)
    }
    asm volatile("s_wait_dscnt 0" ::: "memory");

    // ---- Accumulate logits partial: h_slice(16x16) @ Wcls[16n:16n+16,:] ----
#pragma unroll
    for (int kb = 0; kb < 4; ++kb) {
      int m  = lane & 15;
      int k0 = (kb << 2) + ((lane >> 4) << 1);
      v2f aH;
      aH.x = sHt[m * 20 + k0];
      aH.y = sHt[m * 20 + k0 + 1];
      v2f bC;
      bC.x = Wcls[(n0 + k0) * CCLS + nlane];
      bC.y = Wcls[(n0 + k0 + 1) * CCLS + nlane];
      logacc = wmma4(aH, bC, logacc);
    }
    asm volatile("s_wait_dscnt 0" ::: "memory");  // guard WAR on sHt
  }

  // ---- Store logits tile (C=16 == one WMMA tile width) ----
#pragma unroll
  for (int r = 0; r < 8; ++r) {
    long g = r0 + r + mhalf;
    logits[g * CCLS + nlane] = logacc[r];
  }
}

// ---------------------------------------------------------------------------
// Kernel 2: CRF log-likelihood.  One wave32 per batch element.
// ---------------------------------------------------------------------------
__global__ __launch_bounds__(32) void crf_kernel(
    const float* __restrict__ logits,  // (B, L, 16)
    const float* __restrict__ T,       // (16, 16)
    const int* __restrict__ tags,      // (B, L)
    const int* __restrict__ seqlen,    // (B,)
    float* __restrict__ ll)            // (B,)
{
  __shared__ float sA[16];
  const int b    = blockIdx.x;
  const int lane = threadIdx.x;
  const int sl   = seqlen[b];
  const int* tg  = tags + (long)b * LL;
  const float* lg = logits + (long)b * LL * CCLS;

  // ---- Gold score: masked unary + transition sums (lane-strided) ----
  float gold = 0.0f;
  for (int l = lane; l < LL; l += 32) {
    if (l < sl) {
      int t = tg[l];
      gold += lg[l * CCLS + t];
      if (l >= 1) gold += T[tg[l - 1] * CCLS + t];
    }
  }
#pragma unroll
  for (int off = 16; off > 0; off >>= 1) gold += __shfl_down(gold, off, 32);

  // ---- Forward (alpha) recurrence: lanes 0..15 own CRF states ----
  float tj[16];
  if (lane < 16) {
#pragma unroll
    for (int i = 0; i < 16; ++i) tj[i] = T[i * CCLS + lane];  // column j=lane
    sA[lane] = lg[lane];                                      // alpha0
  }
  asm volatile("s_wait_dscnt 0" ::: "memory");

  for (int l = 1; l < LL; ++l) {
    float nv = 0.0f;
    if (lane < 16) {
      float mx = -INFINITY;
#pragma unroll
      for (int i = 0; i < 16; ++i) mx = fmaxf(mx, sA[i] + tj[i]);
      float s = 0.0f;
#pragma unroll
      for (int i = 0; i < 16; ++i) s += __expf(sA[i] + tj[i] - mx);
      nv = mx + __logf(s) + lg[l * CCLS + lane];
    }
    // All reads of sA for this step precede the write (in-order DS, one wave).
    if (lane < 16 && l < sl) sA[lane] = nv;
    asm volatile("s_wait_dscnt 0" ::: "memory");
  }

  if (lane == 0) {
    float mx = -INFINITY;
#pragma unroll
    for (int i = 0; i < 16; ++i) mx = fmaxf(mx, sA[i]);
    float s = 0.0f;
#pragma unroll
    for (int i = 0; i < 16; ++i) s += __expf(sA[i] - mx);
    ll[b] = gold - (mx + __logf(s));
  }
}

// ---------------------------------------------------------------------------
extern "C" void kernel_launch(void* const* d_in, const int* in_sizes, int n_in,
                              void* d_out, int out_size, void* d_ws,
                              size_t ws_size, hipStream_t stream) {
  const float* trip   = (const float*)d_in[0];  // (B, L, 128)
  const float* region = (const float*)d_in[1];  // (B, L+1, 128)
  const float* W      = (const float*)d_in[2];  // (128, 512)
  const float* Uiou   = (const float*)d_in[3];  // (128, 384)
  const float* Uf     = (const float*)d_in[4];  // (128, 128)
  const float* cc     = (const float*)d_in[5];  // (2, 128)
  const float* Wcls   = (const float*)d_in[6];  // (128, 16)
  const float* T      = (const float*)d_in[7];  // (16, 16)
  const int* tags     = (const int*)d_in[8];    // (B, L)
  const int* slen     = (const int*)d_in[9];    // (B,)

  float* logits = (float*)d_out;                  // B*L*16 floats
  float* ll     = logits + (long)BB * LL * CCLS;  // then B floats

  // 8192 16-row tiles, one wave32 per block (TDM descriptors stay uniform).
  const int tiles = (BB * LL) / 16;
  tree_lstm_logits_kernel<<<tiles, 32, 0, stream>>>(trip, region, W, Uiou, Uf,
                                                    cc, Wcls, logits);
  crf_kernel<<<BB, 32, 0, stream>>>(logits, T, tags, slen, ll);
}